// NeuroLMTokenizer_85435489452289
// MI455X (gfx1250) — compile-verified
//
#include <hip/hip_runtime.h>
#include <hip/hip_bf16.h>

// ---------------- WMMA helpers (CDNA5 f32 16x16x4) ----------------
typedef float v2f __attribute__((ext_vector_type(2)));
typedef float v8f __attribute__((ext_vector_type(8)));

__device__ inline v8f wmma4(v2f a, v2f b, v8f c) {
    // D = A(16x4) * B(4x16) + C(16x16), all f32
    return __builtin_amdgcn_wmma_f32_16x16x4_f32(false, a, false, b, (short)0, c, false, false);
}

__device__ inline float gelu_exact(float x) {
    return 0.5f * x * (1.0f + erff(x * 0.70710678118654752440f));
}

__device__ inline unsigned long long shfl_xor_u64(unsigned long long v, int m) {
    unsigned lo = (unsigned)v, hi = (unsigned)(v >> 32);
    lo = __shfl_xor(lo, m, 32);
    hi = __shfl_xor(hi, m, 32);
    return ((unsigned long long)hi << 32) | lo;
}

// ---------------- problem constants ----------------
#define MROWS   16384
#define DDIM    512
#define EDIM    128
#define NEMB    8192
#define FDIM    400

// workspace layout (float offsets)
#define O_X1       ((size_t)0)          // 6553600  conv buffers (ping)
#define O_X2       ((size_t)6553600)    // 6553600  conv buffers (pong) / AP_PROJ alias
#define O_BP_PROJ  ((size_t)13107200)   // 204800
#define O_AP_FEAT  ((size_t)13312000)   // 8388608  / AP_Q alias
#define O_BP_PREQ  ((size_t)21700608)   // 65536
#define O_ZQ       ((size_t)21766144)   // 2097152
#define O_AP_ZN    ((size_t)23863296)   // 2097152  (zn stored packed)
#define O_ZN2      ((size_t)25960448)   // 16384
#define O_BP_EN    ((size_t)25976832)   // 1048576  (normalized codebook, packed)
#define O_EN2      ((size_t)27025408)   // 8192
#define O_BEST     ((size_t)27033600)   // 32768 floats = 16384 u64
#define O_BP_POSTQ ((size_t)27066368)   // 65536
#define O_VQACC    ((size_t)27131904)   // 1

// d_out layout (floats): tokens | features_q | features | vq_loss
#define OUT_TOKENS 0
#define OUT_FEATQ  16384
#define OUT_FEAT   8404992
#define OUT_VQ     16793600

// packed operand addressing: element (r,k) of a row-major [R x K] matrix lives at
// ((r>>4)*(K/4) + (k>>2))*64 + (16*((k>>1)&1) + (r&15))*2 + (k&1)
__device__ inline size_t packed_addr(int r, int k, int Ksteps) {
    return ((size_t)(r >> 4) * Ksteps + (k >> 2)) * 64 + (size_t)((16 * ((k >> 1) & 1) + (r & 15)) * 2 + (k & 1));
}

// ---------------- init ----------------
__global__ void k_init(unsigned long long* best, float* vqacc) {
    int i = blockIdx.x * blockDim.x + threadIdx.x;
    if (i < MROWS) best[i] = ~0ull;
    if (i == 0) vqacc[0] = 0.f;
}

// ---------------- conv1 + GN + GELU (one block per (batch,group)) ----------------
__global__ void k_conv1(const float* __restrict__ eeg, const float* __restrict__ w,
                        const float* __restrict__ bias, const float* __restrict__ gw,
                        const float* __restrict__ gb, float* __restrict__ out) {
    int b = blockIdx.x >> 2, g = blockIdx.x & 3;
    int tid = threadIdx.x;
    int cl = tid >> 6, n = tid & 63, c = g * 4 + cl;
    __shared__ float sbuf[6400];
    __shared__ float sred[256];
    __shared__ float smv[2];
    const float* er = eeg + ((size_t)b * 64 + n) * 200;
    const float* wr = w + c * 15;
    float acc[25];
#pragma unroll
    for (int t = 0; t < 25; ++t) acc[t] = 0.f;
    for (int k = 0; k < 15; ++k) {
        float wk = wr[k];
#pragma unroll
        for (int t = 0; t < 25; ++t) {
            int p = t * 8 - 7 + k;
            if (p >= 0 && p < 200) acc[t] += wk * er[p];
        }
    }
    float bv = bias[c], ls = 0.f, lq = 0.f;
#pragma unroll
    for (int t = 0; t < 25; ++t) {
        float v = acc[t] + bv;
        sbuf[tid * 25 + t] = v;
        ls += v; lq += v * v;
    }
    sred[tid] = ls; __syncthreads();
    for (int s = 128; s > 0; s >>= 1) { if (tid < s) sred[tid] += sred[tid + s]; __syncthreads(); }
    if (tid == 0) smv[0] = sred[0] * (1.f / 6400.f);
    __syncthreads();
    sred[tid] = lq; __syncthreads();
    for (int s = 128; s > 0; s >>= 1) { if (tid < s) sred[tid] += sred[tid + s]; __syncthreads(); }
    if (tid == 0) { float m = smv[0]; smv[1] = sred[0] * (1.f / 6400.f) - m * m; }
    __syncthreads();
    float m = smv[0], inv = rsqrtf(smv[1] + 1e-5f);
    float gwc = gw[c], gbc = gb[c];
    float* orow = out + (((size_t)b * 16 + c) * 64 + n) * 25;
#pragma unroll
    for (int t = 0; t < 25; ++t) {
        float y = (sbuf[tid * 25 + t] - m) * inv * gwc + gbc;
        orow[t] = gelu_exact(y);
    }
}

// ---------------- conv (1x3, pad 1) + GN + GELU ----------------
__global__ void k_conv3x(const float* __restrict__ in, const float* __restrict__ w,
                         const float* __restrict__ bias, const float* __restrict__ gw,
                         const float* __restrict__ gb, float* __restrict__ out) {
    int b = blockIdx.x >> 2, g = blockIdx.x & 3;
    int tid = threadIdx.x;
    int cl = tid >> 6, n = tid & 63, c = g * 4 + cl;
    __shared__ float sbuf[6400];
    __shared__ float sred[256];
    __shared__ float smv[2];
    float acc[25];
#pragma unroll
    for (int t = 0; t < 25; ++t) acc[t] = 0.f;
    for (int ci = 0; ci < 16; ++ci) {
        const float* ir = in + (((size_t)b * 16 + ci) * 64 + n) * 25;
        const float* wp = w + (c * 16 + ci) * 3;
        float w0 = wp[0], w1 = wp[1], w2 = wp[2];
        float prev = 0.f, cur = ir[0];
#pragma unroll
        for (int t = 0; t < 25; ++t) {
            float nxt = (t < 24) ? ir[t + 1] : 0.f;
            acc[t] += w0 * prev + w1 * cur + w2 * nxt;
            prev = cur; cur = nxt;
        }
    }
    float bv = bias[c], ls = 0.f, lq = 0.f;
#pragma unroll
    for (int t = 0; t < 25; ++t) {
        float v = acc[t] + bv;
        sbuf[tid * 25 + t] = v;
        ls += v; lq += v * v;
    }
    sred[tid] = ls; __syncthreads();
    for (int s = 128; s > 0; s >>= 1) { if (tid < s) sred[tid] += sred[tid + s]; __syncthreads(); }
    if (tid == 0) smv[0] = sred[0] * (1.f / 6400.f);
    __syncthreads();
    sred[tid] = lq; __syncthreads();
    for (int s = 128; s > 0; s >>= 1) { if (tid < s) sred[tid] += sred[tid + s]; __syncthreads(); }
    if (tid == 0) { float m = smv[0]; smv[1] = sred[0] * (1.f / 6400.f) - m * m; }
    __syncthreads();
    float m = smv[0], inv = rsqrtf(smv[1] + 1e-5f);
    float gwc = gw[c], gbc = gb[c];
    float* orow = out + (((size_t)b * 16 + c) * 64 + n) * 25;
#pragma unroll
    for (int t = 0; t < 25; ++t) {
        float y = (sbuf[tid * 25 + t] - m) * inv * gwc + gbc;
        orow[t] = gelu_exact(y);
    }
}

// ---------------- operand repacking into WMMA-native layout ----------------
__global__ void k_pack_rowmajor(const float* __restrict__ src, float* __restrict__ dst,
                                int R, int K, int total) {
    int o = blockIdx.x * blockDim.x + threadIdx.x;
    if (o >= total) return;
    int Ks = K >> 2;
    int j = o & 1, lane = (o >> 1) & 31;
    int ks = (o >> 6) % Ks, rt = o / (64 * Ks);
    int r = rt * 16 + (lane & 15);
    int k = ks * 4 + ((lane >> 4) << 1) + j;
    dst[o] = src[(size_t)r * K + k];
}

// pack x3 (B,16,64,25) -> A-packed [16384 x 400], feature f = t*16 + c
__global__ void k_pack_x3(const float* __restrict__ x3, float* __restrict__ dst) {
    int o = blockIdx.x * blockDim.x + threadIdx.x;
    if (o >= MROWS * FDIM) return;
    int j = o & 1, lane = (o >> 1) & 31;
    int ks = (o >> 6) % 100, rt = o / 6400;
    int r = rt * 16 + (lane & 15);
    int k = ks * 4 + ((lane >> 4) << 1) + j;   // k in [0,400)
    int b = r >> 6, nseq = r & 63, t = k >> 4, c = k & 15;
    dst[o] = x3[(((size_t)b * 16 + c) * 64 + nseq) * 25 + t];
}

// ---------------- GEMM: proj + bias + gelu + pos/chan tables ----------------
__global__ void k_gemm_proj(const float* __restrict__ Ap, const float* __restrict__ Bp,
                            const float* __restrict__ pb, const float* __restrict__ pos,
                            const float* __restrict__ chan, const int* __restrict__ chan_ids,
                            float* __restrict__ feat_out) {
    int mt = blockIdx.x, nt = blockIdx.y, lane = threadIdx.x;
    v8f acc = {0.f, 0.f, 0.f, 0.f, 0.f, 0.f, 0.f, 0.f};
    const float* ap = Ap + (size_t)mt * 100 * 64 + lane * 2;
    const float* bp = Bp + (size_t)nt * 100 * 64 + lane * 2;
    for (int ks = 0; ks < 100; ++ks) {
        v2f a = *(const v2f*)(ap); ap += 64;
        v2f b = *(const v2f*)(bp); bp += 64;
        acc = wmma4(a, b, acc);
    }
    int h = lane >> 4, col = nt * 16 + (lane & 15);
    float bias = pb[col];
#pragma unroll
    for (int v = 0; v < 8; ++v) {
        int row = mt * 16 + v + 8 * h;
        int seq = row & 63;
        float x = gelu_exact(acc[v] + bias);
        x += pos[(size_t)seq * DDIM + col] + chan[(size_t)chan_ids[seq] * DDIM + col];
        feat_out[(size_t)row * DDIM + col] = x;
    }
}

// ---------------- GEMM: A(16384xK) * W(NxK)^T + bias ----------------
__global__ void k_gemm_bias(const float* __restrict__ Ap, const float* __restrict__ Bp,
                            const float* __restrict__ bias, float* __restrict__ out,
                            int Ksteps, int N) {
    int mt = blockIdx.x, nt = blockIdx.y, lane = threadIdx.x;
    v8f acc = {0.f, 0.f, 0.f, 0.f, 0.f, 0.f, 0.f, 0.f};
    const float* ap = Ap + (size_t)mt * Ksteps * 64 + lane * 2;
    const float* bp = Bp + (size_t)nt * Ksteps * 64 + lane * 2;
    for (int ks = 0; ks < Ksteps; ++ks) {
        v2f a = *(const v2f*)(ap); ap += 64;
        v2f b = *(const v2f*)(bp); bp += 64;
        acc = wmma4(a, b, acc);
    }
    int h = lane >> 4, col = nt * 16 + (lane & 15);
    float bv = bias[col];
#pragma unroll
    for (int v = 0; v < 8; ++v) {
        int row = mt * 16 + v + 8 * h;
        out[(size_t)row * N + col] = acc[v] + bv;
    }
}

// ---------------- row L2-normalize (K=128) and write packed + sumsq ----------------
__global__ void k_l2norm_pack128(const float* __restrict__ src, float* __restrict__ dstPacked,
                                 float* __restrict__ sumsq) {
    int row = blockIdx.x * 8 + (threadIdx.x >> 5);
    int lane = threadIdx.x & 31;
    const float4 v = *(const float4*)(src + (size_t)row * EDIM + lane * 4);
    float s = v.x * v.x + v.y * v.y + v.z * v.z + v.w * v.w;
    for (int m = 1; m < 32; m <<= 1) s += __shfl_xor(s, m, 32);
    float inv = 1.f / fmaxf(sqrtf(s), 1e-12f);
    float e[4] = {v.x * inv, v.y * inv, v.z * inv, v.w * inv};
    float s2 = e[0] * e[0] + e[1] * e[1] + e[2] * e[2] + e[3] * e[3];
    for (int m = 1; m < 32; m <<= 1) s2 += __shfl_xor(s2, m, 32);
#pragma unroll
    for (int q = 0; q < 4; ++q) {
        int k = lane * 4 + q;
        dstPacked[packed_addr(row, k, 32)] = e[q];
    }
    if (lane == 0) sumsq[row] = s2;
}

// ---------------- codebook distance GEMM + fused argmin ----------------
// Register-blocked 2 M-tiles x 4 N-tiles per wave: 8 independent WMMA accumulator
// chains (hides XDL latency), 6 b64 loads per 8 WMMAs (0.75 loads/wmma, all L2-resident).
__global__ void k_cbdist(const float* __restrict__ Ap, const float* __restrict__ Bp,
                         const float* __restrict__ zn2, const float* __restrict__ en2,
                         unsigned long long* __restrict__ best) {
    int mt0 = blockIdx.x * 2, nt0 = blockIdx.y * 4, lane = threadIdx.x;
    v8f acc[2][4];
#pragma unroll
    for (int u = 0; u < 2; ++u)
#pragma unroll
        for (int i = 0; i < 4; ++i)
            acc[u][i] = (v8f){0.f, 0.f, 0.f, 0.f, 0.f, 0.f, 0.f, 0.f};
    const float* ap0 = Ap + (size_t)mt0 * 2048 + lane * 2;   // 2048 = 32 ksteps * 64
    const float* ap1 = ap0 + 2048;
    const float* bp  = Bp + (size_t)nt0 * 2048 + lane * 2;
    for (int ks = 0; ks < 32; ++ks) {
        v2f a0 = *(const v2f*)(ap0 + (size_t)ks * 64);
        v2f a1 = *(const v2f*)(ap1 + (size_t)ks * 64);
        v2f b0 = *(const v2f*)(bp + (size_t)(ks * 64));
        v2f b1 = *(const v2f*)(bp + (size_t)(2048 + ks * 64));
        v2f b2 = *(const v2f*)(bp + (size_t)(4096 + ks * 64));
        v2f b3 = *(const v2f*)(bp + (size_t)(6144 + ks * 64));
        acc[0][0] = wmma4(a0, b0, acc[0][0]);
        acc[1][0] = wmma4(a1, b0, acc[1][0]);
        acc[0][1] = wmma4(a0, b1, acc[0][1]);
        acc[1][1] = wmma4(a1, b1, acc[1][1]);
        acc[0][2] = wmma4(a0, b2, acc[0][2]);
        acc[1][2] = wmma4(a1, b2, acc[1][2]);
        acc[0][3] = wmma4(a0, b3, acc[0][3]);
        acc[1][3] = wmma4(a1, b3, acc[1][3]);
    }
    int h = lane >> 4, cbase = lane & 15;
#pragma unroll
    for (int u = 0; u < 2; ++u) {
        unsigned long long key[8];
#pragma unroll
        for (int v = 0; v < 8; ++v) {
            int row = (mt0 + u) * 16 + v + 8 * h;
            float zz = zn2[row];
            unsigned long long kb = ~0ull;
#pragma unroll
            for (int i = 0; i < 4; ++i) {
                int n = (nt0 + i) * 16 + cbase;
                float d = fmaxf(zz + en2[n] - 2.f * acc[u][i][v], 0.f);
                unsigned long long kk = ((unsigned long long)__float_as_uint(d) << 32) | (unsigned)n;
                kb = kk < kb ? kk : kb;
            }
            key[v] = kb;
        }
        for (int m = 1; m < 16; m <<= 1) {
#pragma unroll
            for (int v = 0; v < 8; ++v) {
                unsigned long long o = shfl_xor_u64(key[v], m);
                key[v] = o < key[v] ? o : key[v];
            }
        }
        if ((lane & 15) == 0) {
#pragma unroll
            for (int v = 0; v < 8; ++v)
                atomicMin(&best[(mt0 + u) * 16 + v + 8 * h], key[v]);
        }
    }
}

// ---------------- gather quantized rows (en[idx]) directly into A-packed layout ----------------
__global__ void k_gatherq(const unsigned long long* __restrict__ best,
                          const float* __restrict__ BpEn, float* __restrict__ ApQ) {
    int o = blockIdx.x * blockDim.x + threadIdx.x;
    if (o >= MROWS * EDIM) return;
    int j = o & 1, lane = (o >> 1) & 31;
    int ks = (o >> 6) & 31, mt = o >> 11;
    int row = mt * 16 + (lane & 15);
    int k = ks * 4 + ((lane >> 4) << 1) + j;
    unsigned idx = (unsigned)(best[row] & 0xffffffffu);
    ApQ[o] = BpEn[packed_addr((int)idx, k, 32)];
}

// ---------------- tokens + vq-loss accumulation ----------------
__global__ void k_finalize(const unsigned long long* __restrict__ best,
                           const float* __restrict__ ApZn, const float* __restrict__ ApQ,
                           float* __restrict__ vqacc, float* __restrict__ tokens) {
    int row = blockIdx.x * 8 + (threadIdx.x >> 5);
    int lane = threadIdx.x & 31;
    float s = 0.f;
#pragma unroll
    for (int q = 0; q < 4; ++q) {
        int k = lane * 4 + q;
        size_t a = packed_addr(row, k, 32);
        float d = ApQ[a] - ApZn[a];
        s += d * d;
    }
    for (int m = 1; m < 32; m <<= 1) s += __shfl_xor(s, m, 32);
    if (lane == 0) {
        atomicAdd(vqacc, s);
        tokens[row] = (float)(unsigned)(best[row] & 0xffffffffu);
    }
}

__global__ void k_vqscalar(const float* __restrict__ vqacc, float* __restrict__ out) {
    // vq_loss = q_latent + e_latent = 2 * mean((quantized - zn)^2)
    out[OUT_VQ] = 2.f * vqacc[0] / (float)(MROWS * EDIM);
}

// ---------------- launch ----------------
extern "C" void kernel_launch(void* const* d_in, const int* in_sizes, int n_in,
                              void* d_out, int out_size, void* d_ws, size_t ws_size,
                              hipStream_t stream) {
    const float* eeg      = (const float*)d_in[0];
    const int*   chan_ids = (const int*)d_in[1];
    const float* conv1_w  = (const float*)d_in[2];
    const float* conv1_b  = (const float*)d_in[3];
    const float* gn1_w    = (const float*)d_in[4];
    const float* gn1_b    = (const float*)d_in[5];
    const float* conv2_w  = (const float*)d_in[6];
    const float* conv2_b  = (const float*)d_in[7];
    const float* gn2_w    = (const float*)d_in[8];
    const float* gn2_b    = (const float*)d_in[9];
    const float* conv3_w  = (const float*)d_in[10];
    const float* conv3_b  = (const float*)d_in[11];
    const float* gn3_w    = (const float*)d_in[12];
    const float* gn3_b    = (const float*)d_in[13];
    const float* proj_w   = (const float*)d_in[14];
    const float* proj_b   = (const float*)d_in[15];
    const float* pos_tab  = (const float*)d_in[16];
    const float* chan_tab = (const float*)d_in[17];
    const float* preq_w   = (const float*)d_in[18];
    const float* preq_b   = (const float*)d_in[19];
    const float* postq_w  = (const float*)d_in[20];
    const float* postq_b  = (const float*)d_in[21];
    const float* codebook = (const float*)d_in[22];

    float* ws  = (float*)d_ws;
    float* out = (float*)d_out;
    unsigned long long* best = (unsigned long long*)(ws + O_BEST);
    float* vqacc = ws + O_VQACC;

    k_init<<<64, 256, 0, stream>>>(best, vqacc);

    // conv stack (fused conv + groupnorm + gelu), ping-pong X1/X2
    k_conv1<<<1024, 256, 0, stream>>>(eeg, conv1_w, conv1_b, gn1_w, gn1_b, ws + O_X1);
    k_conv3x<<<1024, 256, 0, stream>>>(ws + O_X1, conv2_w, conv2_b, gn2_w, gn2_b, ws + O_X2);
    k_conv3x<<<1024, 256, 0, stream>>>(ws + O_X2, conv3_w, conv3_b, gn3_w, gn3_b, ws + O_X1);

    // proj GEMM: M=16384, K=400, N=512 (AP_PROJ aliases X2)
    k_pack_x3<<<(MROWS * FDIM + 255) / 256, 256, 0, stream>>>(ws + O_X1, ws + O_X2);
    k_pack_rowmajor<<<(512 * 400 + 255) / 256, 256, 0, stream>>>(proj_w, ws + O_BP_PROJ, 512, 400, 512 * 400);
    k_gemm_proj<<<dim3(1024, 32), 32, 0, stream>>>(ws + O_X2, ws + O_BP_PROJ, proj_b,
                                                   pos_tab, chan_tab, chan_ids, out + OUT_FEAT);

    // preq GEMM: M=16384, K=512, N=128
    k_pack_rowmajor<<<(MROWS * DDIM + 255) / 256, 256, 0, stream>>>(out + OUT_FEAT, ws + O_AP_FEAT,
                                                                    MROWS, DDIM, MROWS * DDIM);
    k_pack_rowmajor<<<(EDIM * DDIM + 255) / 256, 256, 0, stream>>>(preq_w, ws + O_BP_PREQ,
                                                                   EDIM, DDIM, EDIM * DDIM);
    k_gemm_bias<<<dim3(1024, 8), 32, 0, stream>>>(ws + O_AP_FEAT, ws + O_BP_PREQ, preq_b,
                                                  ws + O_ZQ, 128, EDIM);

    // L2 normalize zq rows (-> packed zn) and codebook rows (-> packed en)
    k_l2norm_pack128<<<2048, 256, 0, stream>>>(ws + O_ZQ, ws + O_AP_ZN, ws + O_ZN2);
    k_l2norm_pack128<<<1024, 256, 0, stream>>>(codebook, ws + O_BP_EN, ws + O_EN2);

    // codebook distance GEMM (16384 x 8192 x 128) + fused argmin, 2x4 register-blocked
    k_cbdist<<<dim3(512, 128), 32, 0, stream>>>(ws + O_AP_ZN, ws + O_BP_EN,
                                                ws + O_ZN2, ws + O_EN2, best);

    // gather quantized rows (packed), postq GEMM: M=16384, K=128, N=512
    k_gatherq<<<(MROWS * EDIM + 255) / 256, 256, 0, stream>>>(best, ws + O_BP_EN, ws + O_AP_FEAT);
    k_pack_rowmajor<<<(DDIM * EDIM + 255) / 256, 256, 0, stream>>>(postq_w, ws + O_BP_POSTQ,
                                                                   DDIM, EDIM, DDIM * EDIM);
    k_gemm_bias<<<dim3(1024, 32), 32, 0, stream>>>(ws + O_AP_FEAT, ws + O_BP_POSTQ, postq_b,
                                                   out + OUT_FEATQ, 32, DDIM);

    // tokens + vq loss
    k_finalize<<<2048, 256, 0, stream>>>(best, ws + O_AP_ZN, ws + O_AP_FEAT, vqacc, out + OUT_TOKENS);
    k_vqscalar<<<1, 1, 0, stream>>>(vqacc, out);
}